// GNN_8667244003433
// MI455X (gfx1250) — compile-verified
//
#include <hip/hip_runtime.h>

// Problem constants (from reference)
#define NN 100000
#define EE 600000
#define DD 128

typedef __attribute__((ext_vector_type(2))) float v2f;
typedef __attribute__((ext_vector_type(8))) float v8f;

// ---------------------------------------------------------------- utilities
__global__ void zero_f32(float* __restrict__ p, long long n) {
  long long i = (long long)blockIdx.x * blockDim.x + threadIdx.x;
  if (i < n) p[i] = 0.0f;
}

__global__ void deg_count(const int* __restrict__ dst, float* __restrict__ deg) {
  int e = blockIdx.x * blockDim.x + threadIdx.x;
  if (e < EE) atomicAdd(&deg[dst[e]], 1.0f);
}

__global__ void deg_invert(float* __restrict__ deg) {
  int i = blockIdx.x * blockDim.x + threadIdx.x;
  if (i < NN) {
    float d = deg[i];
    deg[i] = (d > 0.0f) ? 1.0f / fmaxf(d, 1.0f) : 0.0f;
  }
}

// One thread per (edge, 4-float chunk): gather h[src], atomic-add into agg[dst].
// h fits in the 192MB L2, so the random gathers are L2-served; the fp32
// atomics resolve at L2 (global_atomic_add_f32).
__global__ void scatter_add(const float* __restrict__ h,
                            const int* __restrict__ src,
                            const int* __restrict__ dst,
                            float* __restrict__ agg) {
  long long idx = (long long)blockIdx.x * blockDim.x + threadIdx.x;
  if (idx >= (long long)EE * 32) return;
  int e = (int)(idx >> 5);
  int c = (int)(idx & 31);
  int s = src[e], d = dst[e];
  const float4 v = *reinterpret_cast<const float4*>(h + (long long)s * DD + c * 4);
  float* o = agg + (long long)d * DD + c * 4;
  atomicAdd(o + 0, v.x);
  atomicAdd(o + 1, v.y);
  atomicAdd(o + 2, v.z);
  atomicAdd(o + 3, v.w);
}

// Column-sum of m[N][128] accumulated into g[128] (JK='sum' + global_add_pool
// commute, so we only ever need the 128-wide column totals).
__global__ void colsum_add(const float* __restrict__ m, float* __restrict__ g) {
  int t = threadIdx.x;  // 0..127 = feature
  float acc = 0.0f;
  for (int row = blockIdx.x; row < NN; row += gridDim.x)
    acc += m[(long long)row * DD + t];
  atomicAdd(&g[t], acc);
}

// --------------------------------------------------- fused SAGE layer (WMMA)
// out[n][j] = BN( (agg[n]*dinv[n]) @ Wl^T + bl + h[n] @ Wr^T )[j], opt. ReLU.
// Block = 256 threads = 8 waves; block owns a 16-row tile, wave w owns output
// cols [16w,16w+16). K=128 -> 32 V_WMMA_F32_16X16X4_F32 per GEMM, 64 per wave.
// out aliases agg (in place): all K reads complete before the post-barrier
// stores, and row tiles partition the matrix across blocks.
__global__ __launch_bounds__(256) void sage_gemm(
    const float* agg,                 // [N,D] neighbor sums (also output!)
    const float* __restrict__ h,      // [N,D] previous features
    const float* __restrict__ dinv,   // [N]
    const float* __restrict__ wl,     // [D,D] row j = output channel
    const float* __restrict__ bl,     // [D]
    const float* __restrict__ wr,     // [D,D]
    const float* __restrict__ bnw,
    const float* __restrict__ bnb,
    float* out, int do_relu) {
  const int tileM = blockIdx.x * 16;
  const int wid   = threadIdx.x >> 5;
  const int lane  = threadIdx.x & 31;
  const int half  = lane >> 4;   // 0: lanes 0-15, 1: lanes 16-31
  const int ln    = lane & 15;
  const int tileN = wid * 16;

  // A-matrix (16x4 f32): lane ln holds row tileM+ln; VGPR0/1 = K{0,1}+2*half
  const int rowA = tileM + ln;
  const float dv = dinv[rowA];
  const float* aggRow = agg + (long long)rowA * DD;
  const float* hRow   = h   + (long long)rowA * DD;
  // B-matrix (4x16 f32): B[k][j] = W[j][k]; lane ln supplies col tileN+ln
  const float* wlRow = wl + (long long)(tileN + ln) * DD;
  const float* wrRow = wr + (long long)(tileN + ln) * DD;

  v8f acc = {0.f, 0.f, 0.f, 0.f, 0.f, 0.f, 0.f, 0.f};

#pragma unroll
  for (int kb = 0; kb < 32; ++kb) {               // agg @ Wl^T (scaled rows)
    const int k = kb * 4 + half * 2;
    float2 av = *reinterpret_cast<const float2*>(aggRow + k);
    float2 bv = *reinterpret_cast<const float2*>(wlRow + k);
    v2f a; a.x = av.x * dv; a.y = av.y * dv;
    v2f b; b.x = bv.x;      b.y = bv.y;
    acc = __builtin_amdgcn_wmma_f32_16x16x4_f32(false, a, false, b,
                                                (short)0, acc, false, false);
  }
#pragma unroll
  for (int kb = 0; kb < 32; ++kb) {               // h @ Wr^T
    const int k = kb * 4 + half * 2;
    float2 av = *reinterpret_cast<const float2*>(hRow + k);
    float2 bv = *reinterpret_cast<const float2*>(wrRow + k);
    v2f a; a.x = av.x; a.y = av.y;
    v2f b; b.x = bv.x; b.y = bv.y;
    acc = __builtin_amdgcn_wmma_f32_16x16x4_f32(false, a, false, b,
                                                (short)0, acc, false, false);
  }

  __syncthreads();  // in-place safety: whole block done reading agg tile

  // Epilogue: bias + eval-BN (running mean 0, var 1) + optional ReLU.
  const float inv_std = 1.0f / sqrtf(1.0f + 1e-5f);
  const int col = tileN + ln;
  const float bias  = bl[col];
  const float scale = inv_std * bnw[col];
  const float shift = bnb[col];
#pragma unroll
  for (int r = 0; r < 8; ++r) {
    // D layout: VGPR r holds row r (lanes 0-15) / row r+8 (lanes 16-31)
    int m = tileM + r + half * 8;
    float val = (acc[r] + bias) * scale + shift;
    if (do_relu) val = fmaxf(val, 0.0f);
    out[(long long)m * DD + col] = val;
  }
}

// ------------------------------------------------------------- MLP head
// g[128] -> relu(fc1) [256] -> relu(fc2) [128] -> fc3 [1]. One block.
__global__ __launch_bounds__(256) void mlp_head(
    const float* __restrict__ g,
    const float* __restrict__ fc1w, const float* __restrict__ fc1b,
    const float* __restrict__ fc2w, const float* __restrict__ fc2b,
    const float* __restrict__ fc3w, const float* __restrict__ fc3b,
    float* __restrict__ out) {
  __shared__ float gs[128];
  __shared__ float t1[256];
  __shared__ float t2[128];
  int t = threadIdx.x;
  if (t < 128) gs[t] = g[t];
  __syncthreads();
  {
    float a = fc1b[t];
    for (int k = 0; k < 128; ++k) a += fc1w[t * 128 + k] * gs[k];
    t1[t] = fmaxf(a, 0.0f);
  }
  __syncthreads();
  if (t < 128) {
    float a = fc2b[t];
    for (int k = 0; k < 256; ++k) a += fc2w[t * 256 + k] * t1[k];
    t2[t] = fmaxf(a, 0.0f);
  }
  __syncthreads();
  if (t == 0) {
    float a = fc3b[0];
    for (int k = 0; k < 128; ++k) a += fc3w[k] * t2[k];
    out[0] = a;
  }
}

// ------------------------------------------------------------- launcher
extern "C" void kernel_launch(void* const* d_in, const int* in_sizes, int n_in,
                              void* d_out, int out_size, void* d_ws, size_t ws_size,
                              hipStream_t stream) {
  const float* x       = (const float*)d_in[0];
  const int*   ei      = (const int*)d_in[1];   // [2, E]
  const int*   src     = ei;
  const int*   dst     = ei + EE;
  const float* lin_l_w = (const float*)d_in[2]; // [L,D,D]
  const float* lin_l_b = (const float*)d_in[3]; // [L,D]
  const float* lin_r_w = (const float*)d_in[4]; // [L,D,D]
  const float* bn_w    = (const float*)d_in[5]; // [L,D]
  const float* bn_b    = (const float*)d_in[6]; // [L,D]
  const float* fc1w    = (const float*)d_in[7];
  const float* fc1b    = (const float*)d_in[8];
  const float* fc2w    = (const float*)d_in[9];
  const float* fc2b    = (const float*)d_in[10];
  const float* fc3w    = (const float*)d_in[11];
  const float* fc3b    = (const float*)d_in[12];

  // Workspace layout (floats): deg[N] | g[128] | bufA[N*D] | bufB[N*D]
  float* ws   = (float*)d_ws;
  float* deg  = ws;                       // degree -> deg_inv (in place)
  float* g    = ws + NN;                  // 128-wide pooled accumulator
  float* bufA = g + 128;
  float* bufB = bufA + (size_t)NN * DD;

  const long long ND = (long long)NN * DD;

  // deg and g are contiguous: zero both in one launch (ws is poisoned 0xAA).
  zero_f32<<<(NN + 128 + 255) / 256, 256, 0, stream>>>(deg, NN + 128);
  deg_count<<<(EE + 255) / 256, 256, 0, stream>>>(dst, deg);
  deg_invert<<<(NN + 255) / 256, 256, 0, stream>>>(deg);

  colsum_add<<<512, 128, 0, stream>>>(x, g);  // node_rep starts at x

  const float* h = x;
  for (int l = 0; l < 4; ++l) {
    float* agg = (l & 1) ? bufB : bufA;  // ping-pong; GEMM writes in place
    zero_f32<<<(int)((ND + 255) / 256), 256, 0, stream>>>(agg, ND);
    scatter_add<<<(int)(((long long)EE * 32 + 255) / 256), 256, 0, stream>>>(
        h, src, dst, agg);
    sage_gemm<<<NN / 16, 256, 0, stream>>>(
        agg, h, deg,
        lin_l_w + (size_t)l * DD * DD, lin_l_b + l * DD,
        lin_r_w + (size_t)l * DD * DD,
        bn_w + l * DD, bn_b + l * DD,
        agg, (l < 3) ? 1 : 0);
    if (l < 3) colsum_add<<<512, 128, 0, stream>>>(agg, g);  // JK sum
    h = agg;
  }

  mlp_head<<<1, 256, 0, stream>>>(g, fc1w, fc1b, fc2w, fc2b, fc3w, fc3b,
                                  (float*)d_out);
}